// Clustering_28389733826971
// MI455X (gfx1250) — compile-verified
//
#include <hip/hip_runtime.h>
#include <math.h>

typedef __attribute__((ext_vector_type(2))) float v2f;
typedef __attribute__((ext_vector_type(8))) float v8f;

#define NC   10
#define DM   512
#define H4   40
#define NH   8
#define DK   64
#define NB   8
#define LL   2048
#define LK   228
#define UU   100

#define PB   116736      /* pooled floats per batch = 512*228 */
#define PT   933888      /* pooled floats total    = 8*PB     */
#define CCF  9338880     /* cc flat = 10*PT                   */
#define SA   516         /* LDS row stride for A tile (bank spread) */

__device__ __forceinline__ float gelu_f(float x) {
    return 0.5f * x * (1.0f + erff(x * 0.70710678118654752f));
}

// ---------------------------------------------------------------------------
// MaxPool1d(kernel=9, stride=9, pad=4, pad_val=-inf) on [b, 512, 2048] flat.
// ---------------------------------------------------------------------------
__global__ void pool_kernel(const float* __restrict__ in, float* __restrict__ out) {
    int o = blockIdx.x * blockDim.x + threadIdx.x;
    if (o >= PT) return;
    int b   = o / PB;
    int rem = o - b * PB;
    int c   = rem / LK;
    int t   = rem - c * LK;
    int start = t * 9 - 4;
    int s0 = start < 0 ? 0 : start;
    int s1 = start + 9; if (s1 > LL) s1 = LL;
    const float* p = in + (size_t)b * (DM * LL) + (size_t)c * LL;
    float m = -INFINITY;
    for (int tt = s0; tt < s1; ++tt) m = fmaxf(m, p[tt]);
    out[o] = m;
}

// ---------------------------------------------------------------------------
// One MLP head: h1 = gelu(A@W1+b1) via WMMA f32 16x16x4, layer2 + softmax.
// A tile in LDS [112][SA] (rows >=100 are zero). lC gets softmax rows [100][12].
// B-fragment loads are unconditional (clamped address * 0/1 mask) so the inner
// loop has no EXEC-mask churn around the WMMAs.
// ---------------------------------------------------------------------------
__device__ void mlp_head(const float* lA, float* lH1, float* lC,
                         const float* __restrict__ W1, const float* __restrict__ b1,
                         const float* __restrict__ W2, const float* __restrict__ b2,
                         int tid)
{
    const int wave = tid >> 5;
    const int lane = tid & 31;
    const int ml   = lane & 15;
    const int kb   = (lane >> 4) << 1;      // 0 or 2

    // 7 M-tiles x 3 N-tiles = 21 tile pairs over 8 waves, K = 512 (128 steps)
    for (int p = wave; p < 21; p += 8) {
        int mt = p / 3, nt = p - mt * 3;
        int n  = nt * 16 + ml;
        bool nv = (n < H4);
        float msk = nv ? 1.0f : 0.0f;
        const float* wcol = W1 + (nv ? n : (H4 - 1));   // always-valid column
        const float* arow = lA + (mt * 16 + ml) * SA;
        v8f acc = {};
        for (int kc = 0; kc < DM / 4; ++kc) {
            int k0 = kc * 4 + kb;
            v2f av = *(const v2f*)(arow + k0);           // ds_load_b64
            v2f bv;
            bv.x = wcol[k0 * H4] * msk;
            bv.y = wcol[(k0 + 1) * H4] * msk;
            acc = __builtin_amdgcn_wmma_f32_16x16x4_f32(false, av, false, bv,
                                                        (short)0, acc, false, false);
        }
        if (nv) {
            float bias = b1[n];
            int mbase = mt * 16 + ((lane >> 4) << 3);
            for (int r = 0; r < 8; ++r)
                lH1[(mbase + r) * 48 + n] = gelu_f(acc[r] + bias);
        }
    }
    __syncthreads();

    // layer 2: [100,40] @ [40,10] + b2
    for (int o = tid; o < UU * NC; o += 256) {
        int u = o / NC, c = o - u * NC;
        const float* hr = lH1 + u * 48;
        float s = b2[c];
        for (int j = 0; j < H4; ++j) s += hr[j] * W2[j * NC + c];
        lC[u * 12 + c] = s;
    }
    __syncthreads();

    // softmax over the 10 clusters
    if (tid < UU) {
        float* row = lC + tid * 12;
        float mx = row[0];
        for (int c = 1; c < NC; ++c) mx = fmaxf(mx, row[c]);
        float s = 0.0f;
        for (int c = 0; c < NC; ++c) { float e = expf(row[c] - mx); row[c] = e; s += e; }
        float inv = 1.0f / s;
        for (int c = 0; c < NC; ++c) row[c] *= inv;
    }
    __syncthreads();
}

// ---------------------------------------------------------------------------
// Per-(b,l_k) fused kernel: gather K_unf tile -> both MLPs -> stats/loss ->
// argmax scatter centers -> cc = gelu(centers@Wp + bp) written flat [10,8,228,512].
// ---------------------------------------------------------------------------
__global__ void cluster_kernel(const float* __restrict__ poolK,
                               const float* __restrict__ Wk1, const float* __restrict__ bk1,
                               const float* __restrict__ Wk2, const float* __restrict__ bk2,
                               const float* __restrict__ Wq1, const float* __restrict__ bq1,
                               const float* __restrict__ Wq2, const float* __restrict__ bq2,
                               const float* __restrict__ Wp,  const float* __restrict__ bp,
                               float* __restrict__ muOut, float* __restrict__ ccflat,
                               float* __restrict__ acc)
{
    extern __shared__ char smem[];
    float* lA  = (float*)smem;           // 112*516
    float* lH1 = lA  + 112 * SA;         // 112*48
    float* lCk = lH1 + 112 * 48;         // 100*12
    float* lCq = lCk + 100 * 12;         // 100*12
    float* lS  = lCq + 100 * 12;         // 100
    float* lT  = lS  + 100;              // 16 (10 used)
    float* red = lT  + 16;               // 256

    const int tid = threadIdx.x;
    const int b   = blockIdx.x / LK;
    const int lk  = blockIdx.x - b * LK;

    // Gather the K_unf tile (torch unfold + transpose + reshape scramble).
    for (int idx = tid; idx < 112 * DM; idx += 256) {
        int m = idx >> 9;
        int c = idx & 511;
        float v = 0.0f;
        if (m < UU) {
            unsigned flat = (unsigned)lk * 51200u + (unsigned)m * 512u + (unsigned)c;
            unsigned h  = flat / 1459200u;
            unsigned r1 = flat - h * 1459200u;
            unsigned l2 = r1 / 6400u;
            unsigned r2 = r1 - l2 * 6400u;
            unsigned dk = r2 / 100u;
            unsigned up = r2 - dk * 100u;
            unsigned j  = (unsigned)b + up;     // index into zero-padded K stack
            if (j >= 100u)
                v = poolK[(j - 99u) * (unsigned)PB + h * 14592u + l2 * 64u + dk];
        }
        lA[m * SA + c] = v;
    }
    __syncthreads();

    mlp_head(lA, lH1, lCk, Wk1, bk1, Wk2, bk2, tid);   // ck (softmax)
    mlp_head(lA, lH1, lCq, Wq1, bq1, Wq2, bq2, tid);   // cq (softmax)

    // Per-row stats -> mu (global) and log_prob partial sum.
    float lp = 0.0f;
    if (tid < UU) {
        const float* q = lCq + tid * 12;
        const float* k = lCk + tid * 12;
        float mq = 0.0f, mk = 0.0f;
        for (int c = 0; c < NC; ++c) { mq += q[c]; mk += k[c]; }
        mq *= 0.1f; mk *= 0.1f;
        float var = 0.0f;
        for (int c = 0; c < NC; ++c) { float d = q[c] - mq; var += d * d; }
        var *= (1.0f / 9.0f);                      // unbiased (ddof=1)
        float sigma = log1pf(expf(sqrtf(var)));    // softplus(std)
        float z = (mk - mq) / sigma;
        lp = -0.5f * z * z - logf(sigma) - 0.918938533204672741f; // 0.5*log(2*pi)
        muOut[b * (LK * UU) + lk * UU + tid] = mq;
    }
    red[tid] = lp;
    __syncthreads();
    for (int s = 128; s > 0; s >>= 1) {
        if (tid < s) red[tid] += red[tid + s];
        __syncthreads();
    }
    if (tid == 0) atomicAdd(&acc[0], red[0]);

    // Scatter-style centers: total[c] and S[argmax(u)][c].
    if (tid < 110) { if (tid < 100) lS[tid] = 0.0f; else lT[tid - 100] = 0.0f; }
    __syncthreads();
    if (tid < UU) {
        const float* q = lCq + tid * 12;
        int bi = 0; float bvv = q[0];
        for (int c = 1; c < NC; ++c) if (q[c] > bvv) { bvv = q[c]; bi = c; }
        for (int c = 0; c < NC; ++c) {
            atomicAdd(&lT[c], q[c]);
            atomicAdd(&lS[bi * NC + c], q[c]);
        }
    }
    __syncthreads();

    // cc[i, b, lk, m] = gelu(centers[i,:] @ Wp[:,m] + bp[m])
    for (int o = tid; o < NC * DM; o += 256) {
        int i = o >> 9;
        int m = o & 511;
        float s = bp[m];
        for (int c = 0; c < NC; ++c) {
            float cen = (lT[c] - lS[i * NC + c]) * 0.01f;   // /unfolding
            s += cen * Wp[c * DM + m];
        }
        ccflat[(size_t)i * PT + (size_t)b * PB + (size_t)lk * DM + m] = gelu_f(s);
    }
}

// ---------------------------------------------------------------------------
// ccs[b,h,k,d] = sum_c cc_reshaped[b,h,c,k,d]   (reshape is a flat scramble)
// ---------------------------------------------------------------------------
__global__ void ccs_kernel(const float* __restrict__ ccflat, float* __restrict__ ccs) {
    int o = blockIdx.x * blockDim.x + threadIdx.x;
    if (o >= PT) return;
    int d  = o & 63;
    int k  = (o >> 6) % LK;
    int t  = o / (LK * 64);          // t = b*8 + h
    int b2 = t >> 3;
    int h2 = t & 7;
    size_t base = (size_t)b2 * 1167360 + (size_t)h2 * 145920 + (size_t)k * 64 + d;
    float s = 0.0f;
    for (int c = 0; c < NC; ++c) s += ccflat[base + (size_t)c * 14592];
    ccs[o] = s;
}

// ---------------------------------------------------------------------------
// CE term: per (b,u), log_softmax over l_k of mu, accumulate -sum(mu*logp).
// ---------------------------------------------------------------------------
__global__ void ce_kernel(const float* __restrict__ mu, float* __restrict__ acc) {
    __shared__ float sred[256];
    int p = blockIdx.x;
    int b = p / UU, u = p - b * UU;
    int tid = threadIdx.x;
    float x = (tid < LK) ? mu[b * (LK * UU) + tid * UU + u] : -INFINITY;
    sred[tid] = x;
    __syncthreads();
    for (int s = 128; s > 0; s >>= 1) { if (tid < s) sred[tid] = fmaxf(sred[tid], sred[tid + s]); __syncthreads(); }
    float mx = sred[0];
    __syncthreads();
    float e = (tid < LK) ? expf(x - mx) : 0.0f;
    sred[tid] = e;
    __syncthreads();
    for (int s = 128; s > 0; s >>= 1) { if (tid < s) sred[tid] += sred[tid + s]; __syncthreads(); }
    float lse = mx + logf(sred[0]);
    __syncthreads();
    float v = (tid < LK) ? x * (lse - x) : 0.0f;
    sred[tid] = v;
    __syncthreads();
    for (int s = 128; s > 0; s >>= 1) { if (tid < s) sred[tid] += sred[tid + s]; __syncthreads(); }
    if (tid == 0) atomicAdd(&acc[1], sred[0]);
}

// ---------------------------------------------------------------------------
// Attention: per (b,h, 64-row Q block): scores = Q@ccs^T/8 -> softmax -> @Vp.
// All operands LDS-resident (~205 KB); both GEMMs via WMMA f32 16x16x4.
// ---------------------------------------------------------------------------
__global__ void attn_kernel(const float* __restrict__ Q,
                            const float* __restrict__ ccs,
                            const float* __restrict__ poolV,
                            float* __restrict__ out)
{
    extern __shared__ char smem[];
    float* lC = (float*)smem;            // 240*68  (ccs, k-major)
    float* lV = lC + 240 * 68;           // 240*68
    float* lQ = lV + 240 * 68;           // 64*68
    float* lS = lQ + 64 * 68;            // 64*244

    const int tid = threadIdx.x;
    const int bh  = blockIdx.x >> 5;
    const int q0  = (blockIdx.x & 31) * 64;

    for (int idx = tid; idx < 240 * 64; idx += 256) {
        int k = idx >> 6, d = idx & 63;
        float vc = 0.0f, vv = 0.0f;
        if (k < LK) {
            size_t g = (size_t)bh * 14592 + (size_t)k * 64 + d;
            vc = ccs[g];
            vv = poolV[g];
        }
        lC[k * 68 + d] = vc;
        lV[k * 68 + d] = vv;
    }
    for (int idx = tid; idx < 64 * 64; idx += 256) {
        int m = idx >> 6, d = idx & 63;
        lQ[m * 68 + d] = Q[((size_t)bh * LL + q0 + m) * 64 + d];
    }
    __syncthreads();

    const int wave = tid >> 5, lane = tid & 31;
    const int ml = lane & 15;
    const int kb = (lane >> 4) << 1;

    // scores: 4 M-tiles x 15 N-tiles, K=64
    for (int p = wave; p < 60; p += 8) {
        int mt = p / 15, nt = p - mt * 15;
        const float* arow = lQ + (mt * 16 + ml) * 68;
        const float* brow = lC + (nt * 16 + ml) * 68;
        v8f acc = {};
        for (int kc = 0; kc < 16; ++kc) {
            int k0 = kc * 4 + kb;
            v2f av = *(const v2f*)(arow + k0);   // ds_load_b64
            v2f bv = *(const v2f*)(brow + k0);   // ds_load_b64
            acc = __builtin_amdgcn_wmma_f32_16x16x4_f32(false, av, false, bv,
                                                        (short)0, acc, false, false);
        }
        int mbase = mt * 16 + ((lane >> 4) << 3);
        for (int r = 0; r < 8; ++r)
            lS[(mbase + r) * 244 + nt * 16 + ml] = acc[r] * 0.125f; // 1/sqrt(64)
    }
    __syncthreads();

    if (tid < 64) {
        float* row = lS + tid * 244;
        float mx = -INFINITY;
        for (int k = 0; k < LK; ++k) mx = fmaxf(mx, row[k]);
        float s = 0.0f;
        for (int k = 0; k < LK; ++k) { float e = expf(row[k] - mx); row[k] = e; s += e; }
        float inv = 1.0f / s;
        for (int k = 0; k < LK; ++k) row[k] *= inv;
        for (int k = LK; k < 240; ++k) row[k] = 0.0f;  // kill pad cols
    }
    __syncthreads();

    // context: 4 M-tiles x 4 N-tiles, K=240
    for (int p = wave; p < 16; p += 8) {
        int mt = p >> 2, nt = p & 3;
        const float* arow = lS + (mt * 16 + ml) * 244;
        v8f acc = {};
        for (int kc = 0; kc < 60; ++kc) {
            int k0 = kc * 4 + kb;
            v2f av = *(const v2f*)(arow + k0);   // ds_load_b64
            v2f bv;
            bv.x = lV[k0 * 68 + nt * 16 + ml];
            bv.y = lV[(k0 + 1) * 68 + nt * 16 + ml];
            acc = __builtin_amdgcn_wmma_f32_16x16x4_f32(false, av, false, bv,
                                                        (short)0, acc, false, false);
        }
        int mbase = mt * 16 + ((lane >> 4) << 3);
        for (int r = 0; r < 8; ++r)
            out[((size_t)bh * LL + q0 + mbase + r) * 64 + nt * 16 + ml] = acc[r];
    }
}

__global__ void init_kernel(float* acc) {
    if (threadIdx.x < 2) acc[threadIdx.x] = 0.0f;
}

__global__ void fin_kernel(const float* __restrict__ acc, float* __restrict__ out) {
    out[0] = -acc[0] / 182400.0f + acc[1] / 800.0f;
}

// ---------------------------------------------------------------------------
extern "C" void kernel_launch(void* const* d_in, const int* in_sizes, int n_in,
                              void* d_out, int out_size, void* d_ws, size_t ws_size,
                              hipStream_t stream) {
    (void)in_sizes; (void)n_in; (void)out_size; (void)ws_size;
    const float* Q   = (const float*)d_in[0];
    const float* K   = (const float*)d_in[1];
    const float* V   = (const float*)d_in[2];
    const float* Wk1 = (const float*)d_in[3];
    const float* bk1 = (const float*)d_in[4];
    const float* Wk2 = (const float*)d_in[5];
    const float* bk2 = (const float*)d_in[6];
    const float* Wq1 = (const float*)d_in[7];
    const float* bq1 = (const float*)d_in[8];
    const float* Wq2 = (const float*)d_in[9];
    const float* bq2 = (const float*)d_in[10];
    const float* Wp  = (const float*)d_in[11];
    const float* bp  = (const float*)d_in[12];

    float* ws  = (float*)d_ws;
    float* pK  = ws;                       // PT
    float* pV  = ws + PT;                  // PT
    float* mu  = ws + 2 * (size_t)PT;      // 182400
    float* ccf = mu + (size_t)NB * LK * UU;
    float* ccs = ccf + (size_t)CCF;        // PT
    float* acc = ccs + (size_t)PT;         // 2

    float* ctx     = (float*)d_out;
    float* lossOut = ctx + (size_t)NB * NH * LL * DK;   // 8388608

    init_kernel<<<1, 32, 0, stream>>>(acc);

    int pb = (PT + 255) / 256;
    pool_kernel<<<pb, 256, 0, stream>>>(K, pK);
    pool_kernel<<<pb, 256, 0, stream>>>(V, pV);

    size_t smemB = (size_t)(112 * SA + 112 * 48 + 2 * 100 * 12 + 100 + 16 + 256) * sizeof(float);
    hipFuncSetAttribute(reinterpret_cast<const void*>(cluster_kernel),
                        hipFuncAttributeMaxDynamicSharedMemorySize, (int)smemB);
    cluster_kernel<<<NB * LK, 256, smemB, stream>>>(pK, Wk1, bk1, Wk2, bk2,
                                                    Wq1, bq1, Wq2, bq2, Wp, bp,
                                                    mu, ccf, acc);

    ccs_kernel<<<pb, 256, 0, stream>>>(ccf, ccs);
    ce_kernel<<<NB * UU, 256, 0, stream>>>(mu, acc);

    size_t smemE = (size_t)(2 * 240 * 68 + 64 * 68 + 64 * 244) * sizeof(float);
    hipFuncSetAttribute(reinterpret_cast<const void*>(attn_kernel),
                        hipFuncAttributeMaxDynamicSharedMemorySize, (int)smemE);
    attn_kernel<<<NB * NH * (LL / 64), 256, smemE, stream>>>(Q, ccs, pV, ctx);

    fin_kernel<<<1, 1, 0, stream>>>(acc, lossOut);
}